// SearchAttention_48447231099071
// MI455X (gfx1250) — compile-verified
//
#include <hip/hip_runtime.h>

typedef __attribute__((ext_vector_type(2))) float v2f;
typedef __attribute__((ext_vector_type(8))) float v8f;

#define WMMA_F32(a, b, c) \
  __builtin_amdgcn_wmma_f32_16x16x4_f32(false, (a), false, (b), (short)0, (c), false, false)

constexpr int B_ = 4, H_ = 16, S_ = 1024, D_ = 64, E_ = 1024;
constexpr int QB = 32;            // queries per attention workgroup
constexpr int SROW = 1026;        // padded LDS row stride (floats)
constexpr int MLOFF_W = QB * SROW;            // word offset of M slab in LDS
constexpr int LDS_WORDS = MLOFF_W + QB * S_;  // 65,600 words = 262,400 B
constexpr float INV_SQRT_D = 0.125f;

// ---------------------------------------------------------------------------
// Kernel 0a: zero the per-(b,h,k) prob-max buffer (ws is poisoned by harness)
// ---------------------------------------------------------------------------
__global__ void __launch_bounds__(256) init_pmax_kernel(int* __restrict__ p) {
  p[blockIdx.x * 256 + threadIdx.x] = 0;
}

// ---------------------------------------------------------------------------
// Kernel 0b: Vt[b,h,d,k] = V[b,h,k,d]   (so P@V B-fragments are contiguous)
// ---------------------------------------------------------------------------
__global__ void __launch_bounds__(256)
transv_kernel(const float* __restrict__ V, float* __restrict__ Vt) {
  __shared__ float t[64][65];
  const int bh = blockIdx.x >> 4;
  const int kt = blockIdx.x & 15;                 // 16 k-tiles of 64
  const size_t base = ((size_t)bh * S_ + (size_t)kt * 64) * D_;
  for (int i = threadIdx.x; i < 64 * 64; i += 256)
    t[i >> 6][i & 63] = V[base + i];
  __syncthreads();
  for (int i = threadIdx.x; i < 64 * 64; i += 256) {
    const int dd = i >> 6, kk = i & 63;
    Vt[((size_t)bh * D_ + dd) * S_ + kt * 64 + kk] = t[kk][dd];
  }
}

// ---------------------------------------------------------------------------
// Kernel 1: Q projection  Qh[b,h,q,d] = (X @ Wq^T + bq) reshaped per-head
// 256 thr (8 waves); block tile 64(M) x 128(N); wave tile 16 x 64.
// ---------------------------------------------------------------------------
__global__ void __launch_bounds__(256)
qproj_kernel(const float* __restrict__ X, const float* __restrict__ Wq,
             const float* __restrict__ bq, float* __restrict__ Qh) {
  const int tid = threadIdx.x, lane = tid & 31, w = tid >> 5;
  const int lo = lane & 15, half = lane >> 4;
  const int m0 = blockIdx.x * 64 + (w >> 1) * 16;   // global M row (= b*S + q)
  const int n0 = blockIdx.y * 128 + (w & 1) * 64;   // global N col (= e)

  const float* Ab = X + (size_t)(m0 + lo) * E_ + 2 * half;
  const float* Bb[4];
#pragma unroll
  for (int nt = 0; nt < 4; ++nt)
    Bb[nt] = Wq + (size_t)(n0 + nt * 16 + lo) * E_ + 2 * half;

  v8f acc[4] = {};
  v2f a_c = *(const v2f*)(Ab);
  v2f b_c[4];
#pragma unroll
  for (int nt = 0; nt < 4; ++nt) b_c[nt] = *(const v2f*)(Bb[nt]);

#pragma unroll 4
  for (int ks = 0; ks < E_ / 4; ++ks) {
    const int kn = (ks + 1 < E_ / 4) ? ks + 1 : ks;   // clamped prefetch index
    v2f a_n = *(const v2f*)(Ab + kn * 4);
    v2f b_n[4];
#pragma unroll
    for (int nt = 0; nt < 4; ++nt) b_n[nt] = *(const v2f*)(Bb[nt] + kn * 4);
#pragma unroll
    for (int nt = 0; nt < 4; ++nt) acc[nt] = WMMA_F32(a_c, b_c[nt], acc[nt]);
    a_c = a_n;
#pragma unroll
    for (int nt = 0; nt < 4; ++nt) b_c[nt] = b_n[nt];
  }

#pragma unroll
  for (int nt = 0; nt < 4; ++nt) {
    const int e = n0 + nt * 16 + lo;
    const float bias = bq[e];
    const int h = e >> 6, dd = e & 63;
#pragma unroll
    for (int i = 0; i < 8; ++i) {
      const int m = m0 + i + 8 * half;
      const int b = m >> 10, q = m & 1023;
      Qh[((size_t)(b * H_ + h) * S_ + q) * D_ + dd] = acc[nt][i] + bias;
    }
  }
}

// ---------------------------------------------------------------------------
// Kernel 2: attention for one (b, h, 32-query block). Single pass over M,
// streamed with GLOBAL_LOAD_ASYNC_TO_LDS_B128 concurrently with QK^T WMMAs.
// ---------------------------------------------------------------------------
__global__ void __launch_bounds__(256)
attn_kernel(const float* __restrict__ Qh, const float* __restrict__ Kg,
            const float* __restrict__ Vt, const float* __restrict__ Mg,
            const float* __restrict__ dpm, float* __restrict__ ctx,
            int* __restrict__ Pmax) {
  extern __shared__ float Sl[];                     // [QB x SROW] scores | [QB x S_] M
  const int tid = threadIdx.x, lane = tid & 31, w = tid >> 5;
  const int lo = lane & 15, half = lane >> 4;
  const int qb = blockIdx.x, h = blockIdx.y, b = blockIdx.z;
  const int q0 = qb * QB;
  const size_t bh = (size_t)(b * H_ + h);

  // -------- Phase 0: async-stream this block's M slab (128 KB) into LDS -----
  // Non-temporal: M is 256 MB stream-once data; keep L2 for K/Vt/Qh re-reads.
  {
    const float* Mbase = Mg + (bh * S_ + (size_t)q0) * S_;
#pragma unroll
    for (int i = 0; i < (QB * S_) / (256 * 4); ++i) {   // 32 x 16B per thread
      const int idx = tid + i * 256;                    // 16-byte chunk index
      const unsigned lds_b = (unsigned)(MLOFF_W * 4 + idx * 16);
      const void* gp = (const void*)(Mbase + idx * 4);
      asm volatile("global_load_async_to_lds_b128 %0, %1, off th:TH_LOAD_NT"
                   :: "v"(lds_b), "v"(gp) : "memory");
    }
  }

  // -------- Phase 1: Sl[q][k] = (Q . K^T) * inv_sqrt_d  via WMMA f32 --------
  {
    const float* Qbase = Qh + (bh * S_ + q0) * D_;
    const float* Kbase = Kg + bh * S_ * D_;
#pragma unroll
    for (int mt = 0; mt < QB / 16; ++mt) {
      v2f af[16];                                   // hoisted A (Q) fragments
      const float* ap = Qbase + (size_t)(mt * 16 + lo) * D_ + 2 * half;
#pragma unroll
      for (int ks = 0; ks < 16; ++ks) af[ks] = *(const v2f*)(ap + ks * 4);

      v2f bf[2][16];                                // double-buffered K frags
      {
        const float* bp = Kbase + (size_t)(w * 128 + lo) * D_ + 2 * half;
#pragma unroll
        for (int ks = 0; ks < 16; ++ks) bf[0][ks] = *(const v2f*)(bp + ks * 4);
      }
#pragma unroll
      for (int nt = 0; nt < 8; ++nt) {              // this wave's 128-key strip
        const int cur = nt & 1;
        if (nt < 7) {                               // prefetch next key tile
          const float* bp =
              Kbase + (size_t)(w * 128 + (nt + 1) * 16 + lo) * D_ + 2 * half;
#pragma unroll
          for (int ks = 0; ks < 16; ++ks)
            bf[cur ^ 1][ks] = *(const v2f*)(bp + ks * 4);
        }
        v8f acc0 = {}, acc1 = {};                   // even/odd k-step chains
#pragma unroll
        for (int ks = 0; ks < 16; ks += 2) {
          acc0 = WMMA_F32(af[ks],     bf[cur][ks],     acc0);
          acc1 = WMMA_F32(af[ks + 1], bf[cur][ks + 1], acc1);
        }
        const int col = w * 128 + nt * 16 + lo;
        const int rb = mt * 16 + 8 * half;
#pragma unroll
        for (int i = 0; i < 8; ++i)
          Sl[(rb + i) * SROW + col] = (acc0[i] + acc1[i]) * INV_SQRT_D;
      }
    }
  }
  asm volatile("s_wait_asynccnt 0x0" ::: "memory");  // M slab resident in LDS
  __syncthreads();

  // -------- Phase 2: fused (-M from LDS), row softmax, dp_mask --------------
  for (int r = w; r < QB; r += 8) {
    float* row = Sl + r * SROW;
    const float* Mrow = Sl + MLOFF_W + r * S_;
    float mx = -3.0e38f;
    for (int i = lane; i < S_; i += 32) {
      float t = row[i] - Mrow[i];
      row[i] = t;
      mx = fmaxf(mx, t);
    }
#pragma unroll
    for (int o = 16; o; o >>= 1) mx = fmaxf(mx, __shfl_xor(mx, o, 32));
    float sum = 0.f;
    for (int i = lane; i < S_; i += 32) {
      float e = __expf(row[i] - mx);
      row[i] = e;
      sum += e;
    }
#pragma unroll
    for (int o = 16; o; o >>= 1) sum += __shfl_xor(sum, o, 32);
    const float scale = dpm[b * S_ + q0 + r] / sum;
    for (int i = lane; i < S_; i += 32) row[i] *= scale;
  }
  __syncthreads();

  // -------- Phase 3: per-key max over this q block -> atomic bit-max --------
  for (int k = tid; k < S_; k += 256) {
    float mx = 0.f;                                 // probs >= 0
    for (int q = 0; q < QB; ++q) mx = fmaxf(mx, Sl[q * SROW + k]);
    atomicMax(&Pmax[(b * H_ + h) * S_ + k], __float_as_int(mx));
  }

  // -------- Phase 4: ctx = P @ V  (A from LDS, B from transposed V) ---------
  {
    const int mt = w >> 2;                          // 16-row tile
    const int d0 = (w & 3) * 16;                    // 16-col (head dim) tile
    const float* Ab = Sl + (mt * 16 + lo) * SROW + 2 * half;
    const float* Vb = Vt + (bh * D_ + d0 + lo) * S_ + 2 * half;
    v8f acc0 = {}, acc1 = {};
    v2f a0 = *(const v2f*)(Ab);      v2f v0 = *(const v2f*)(Vb);
    v2f a1 = *(const v2f*)(Ab + 4);  v2f v1 = *(const v2f*)(Vb + 4);
#pragma unroll 4
    for (int ks = 0; ks < S_ / 4; ks += 2) {
      const int k2 = (ks + 2 < S_ / 4) ? ks + 2 : ks;       // clamped prefetch
      const int k3 = (ks + 3 < S_ / 4) ? ks + 3 : ks + 1;
      v2f a2 = *(const v2f*)(Ab + k2 * 4);
      v2f w2 = *(const v2f*)(Vb + k2 * 4);
      v2f a3 = *(const v2f*)(Ab + k3 * 4);
      v2f w3 = *(const v2f*)(Vb + k3 * 4);
      acc0 = WMMA_F32(a0, v0, acc0);
      acc1 = WMMA_F32(a1, v1, acc1);
      a0 = a2; v0 = w2; a1 = a3; v1 = w3;
    }
    const v8f acc = acc0 + acc1;
    const int q = q0 + mt * 16 + 8 * half;
    const int e = h * D_ + d0 + lo;
#pragma unroll
    for (int i = 0; i < 8; ++i)   // write-once output: non-temporal stores
      __builtin_nontemporal_store(acc[i],
                                  ctx + (size_t)(b * S_ + q + i) * E_ + e);
  }
}

// ---------------------------------------------------------------------------
// Kernel 3: target_score[b,k] = sum_h max_q probs
// ---------------------------------------------------------------------------
__global__ void __launch_bounds__(256)
tscore_kernel(const int* __restrict__ Pmax, float* __restrict__ out) {
  const int idx = blockIdx.x * 256 + threadIdx.x;   // 4096 = B*Sk
  const int b = idx >> 10, k = idx & 1023;
  float s = 0.f;
#pragma unroll
  for (int h = 0; h < H_; ++h) s += __int_as_float(Pmax[(b * H_ + h) * S_ + k]);
  out[idx] = s;
}

// ---------------------------------------------------------------------------
extern "C" void kernel_launch(void* const* d_in, const int* in_sizes, int n_in,
                              void* d_out, int out_size, void* d_ws, size_t ws_size,
                              hipStream_t stream) {
  (void)in_sizes; (void)n_in; (void)out_size; (void)ws_size;
  const float* dp_emb  = (const float*)d_in[0];
  const float* dp_mask = (const float*)d_in[1];
  const float* K       = (const float*)d_in[2];
  const float* V       = (const float*)d_in[3];
  const float* M       = (const float*)d_in[4];
  const float* Wq      = (const float*)d_in[5];
  const float* bq      = (const float*)d_in[6];

  float* ctx = (float*)d_out;
  float* ts  = ctx + (size_t)B_ * S_ * E_;

  float* Qh   = (float*)d_ws;                          // 16 MB
  float* Vt   = Qh + (size_t)B_ * H_ * S_ * D_;        // 16 MB
  int*   Pmax = (int*)(Vt + (size_t)B_ * H_ * S_ * D_);// 256 KB

  const int smem = LDS_WORDS * (int)sizeof(float);     // 262,400 B (< 320 KB)
  (void)hipFuncSetAttribute((const void*)attn_kernel,
                            hipFuncAttributeMaxDynamicSharedMemorySize, smem);

  init_pmax_kernel<<<B_ * H_ * S_ / 256, 256, 0, stream>>>(Pmax);
  transv_kernel<<<B_ * H_ * (S_ / 64), 256, 0, stream>>>(V, Vt);
  qproj_kernel<<<dim3(B_ * S_ / 64, E_ / 128), 256, 0, stream>>>(dp_emb, Wq, bq, Qh);
  attn_kernel<<<dim3(S_ / QB, H_, B_), 256, smem, stream>>>(Qh, K, Vt, M, dp_mask, ctx, Pmax);
  tscore_kernel<<<B_ * S_ / 256, 256, 0, stream>>>(Pmax, ts);
}